// MinibatchLayer_80470507258473
// MI455X (gfx1250) — compile-verified
//
#include <hip/hip_runtime.h>
#include <math.h>

#define B_SZ 512
#define D_SZ 2048
#define K_SZ 128
#define P_SZ 5
#define N_SZ (K_SZ * P_SZ)      // 640 output columns of the GEMM
#define OUT_W (D_SZ + K_SZ)     // 2176

typedef __attribute__((ext_vector_type(2))) float v2f;
typedef __attribute__((ext_vector_type(8))) float v8f;
typedef __attribute__((ext_vector_type(4))) unsigned int u32x4;
typedef __attribute__((ext_vector_type(8))) int i32x8;
typedef __attribute__((ext_vector_type(4))) int i32x4;

// ---------------------------------------------------------------------------
// Stage 1: scale[n] = exp(lws[n]) * rsqrt( sum_d theta[d*N + n]^2 )
// ---------------------------------------------------------------------------
__global__ void scale_kernel(const float* __restrict__ theta,
                             const float* __restrict__ lws,
                             float* __restrict__ scale) {
    int n = blockIdx.x * blockDim.x + threadIdx.x;
    if (n >= N_SZ) return;
    float s = 0.f;
#pragma unroll 8
    for (int d = 0; d < D_SZ; ++d) {
        float t = theta[d * N_SZ + n];
        s = fmaf(t, t, s);
    }
    scale[n] = __expf(lws[n]) * rsqrtf(s);
}

// ---------------------------------------------------------------------------
// Stage 2: actv[b, n] = scale[n] * sum_d x[b,d] * theta[d,n]
// fp32 WMMA 16x16x4, one wave per 16x16 tile, 4 waves/block, exact cover.
// ---------------------------------------------------------------------------
__global__ void __launch_bounds__(128)
gemm_wmma_kernel(const float* __restrict__ x,
                 const float* __restrict__ theta,
                 const float* __restrict__ scale,
                 float* __restrict__ actv) {
    const int lane  = threadIdx.x & 31;
    const int wave  = threadIdx.x >> 5;
    const int n0    = (blockIdx.x * 4 + wave) * 16;  // tile origin in N
    const int m0    = blockIdx.y * 16;               // tile origin in M
    const int mr    = lane & 15;                     // A row / B,C column
    const int khalf = (lane >> 4) << 1;              // 0 or 2

    const float* aBase = x + (size_t)(m0 + mr) * D_SZ + khalf;
    const float* bBase = theta + n0 + mr;

    v8f c = {};
#pragma unroll 8
    for (int d = 0; d < D_SZ; d += 4) {
        v2f a = *reinterpret_cast<const v2f*>(aBase + d);
        v2f b;
        b.x = bBase[(size_t)(d + khalf) * N_SZ];
        b.y = bBase[(size_t)(d + khalf + 1) * N_SZ];
        c = __builtin_amdgcn_wmma_f32_16x16x4_f32(
                false, a, false, b, (short)0, c, false, false);
    }

    const float sc = scale[n0 + mr];
    const int mhi  = (lane >> 4) << 3;
#pragma unroll
    for (int r = 0; r < 8; ++r) {
        actv[(size_t)(m0 + mhi + r) * N_SZ + n0 + mr] = c[r] * sc;
    }
}

// ---------------------------------------------------------------------------
// Stage 3: copy x into out[:, 0:2048] (float4 moves).
// ---------------------------------------------------------------------------
__global__ void copy_x_kernel(const float* __restrict__ x,
                              float* __restrict__ out) {
    int idx = blockIdx.x * blockDim.x + threadIdx.x;
    if (idx >= B_SZ * (D_SZ / 4)) return;
    int b = idx / (D_SZ / 4);
    int j = idx % (D_SZ / 4);
    float4 v = reinterpret_cast<const float4*>(x)[idx];
    *reinterpret_cast<float4*>(out + (size_t)b * OUT_W + j * 4) = v;
}

// ---------------------------------------------------------------------------
// Stage 4: per-k pairwise L1 + exp reduction, with the 512x5 activation slice
// DMA'd into LDS by the Tensor Data Mover (single TENSOR_LOAD_TO_LDS per
// block, issued by wave 0; TENSORcnt + workgroup barrier for visibility).
//
// NOTE: the TDM builtin carries no pointer operand, so the compiler cannot
// see that it writes `sa`. Without explicit asm memory clobbers, LLVM folds
// the (apparently never-stored) LDS loads to undef and deletes the compute
// loop — which is exactly what round-2's histogram showed (ds: 14 -> 0).
// The empty `asm volatile("" ::: "memory")` barriers are may-writes to all
// memory and keep the ds_load path alive at zero instruction cost.
// ---------------------------------------------------------------------------
__global__ void __launch_bounds__(512)
pairwise_kernel(const float* __restrict__ actv,
                const float* __restrict__ bias,
                float* __restrict__ out) {
    const int k = blockIdx.x;
    const int b = threadIdx.x;

    __shared__ float sa[B_SZ * P_SZ];   // 10240 B

    // LDS byte offset of sa: low 32 bits of its generic address (aperture map)
    const unsigned ldsOff =
        (unsigned)((unsigned long long)(void*)&sa[0] & 0xFFFFFFFFull);

    if (threadIdx.x < 32) {             // wave 0 issues the TDM descriptor
        unsigned long long gaddr =
            (unsigned long long)(const void*)(actv + (size_t)k * P_SZ);

        u32x4 g0;
        g0[0] = 1u;                                    // count=1 (valid user D#)
        g0[1] = ldsOff;                                // lds_addr
        g0[2] = (unsigned)(gaddr & 0xFFFFFFFFull);     // global_addr[31:0]
        g0[3] = (unsigned)((gaddr >> 32) & 0x01FFFFFFu)
              | (2u << 30);                            // global_addr[56:32], type=2

        i32x8 g1;
        g1[0] = (2 << 16);                             // data_size = 4 bytes
        g1[1] = (int)((N_SZ & 0xFFFF) << 16);          // tensor_dim0[15:0]=640
        g1[2] = (int)((N_SZ >> 16) | ((B_SZ & 0xFFFF) << 16)); // dim0 hi | dim1 lo
        g1[3] = (int)((B_SZ >> 16) | (P_SZ << 16));    // dim1 hi | tile_dim0=5
        g1[4] = B_SZ;                                  // tile_dim1=512, tile_dim2=0
        g1[5] = N_SZ;                                  // tensor_dim0_stride[31:0]
        g1[6] = 0;                                     // stride hi | dim1_stride lo
        g1[7] = 0;

        i32x4 gz4 = {0, 0, 0, 0};                      // groups 2/3 unused (2D)
        i32x8 gz8 = {0, 0, 0, 0, 0, 0, 0, 0};          // extra group (clang-23 ABI)

        __builtin_amdgcn_tensor_load_to_lds(g0, g1, gz4, gz4, gz8, 0);
        __builtin_amdgcn_s_wait_tensorcnt(0);          // s_wait_tensorcnt 0x0
        asm volatile("" ::: "memory");                 // TDM wrote LDS: tell LLVM
    }
    __syncthreads();                                   // LDS visible to all waves
    asm volatile("" ::: "memory");                     // keep ds_loads live

    float my[P_SZ];
#pragma unroll
    for (int p = 0; p < P_SZ; ++p) my[p] = sa[b * P_SZ + p];

    float acc = 0.f;
    for (int b2 = 0; b2 < B_SZ; ++b2) {
        float d = (b2 == b) ? 1.0e6f : 0.0f;           // diagonal -> exp == 0
#pragma unroll
        for (int p = 0; p < P_SZ; ++p)
            d += fabsf(my[p] - sa[b2 * P_SZ + p]);
        acc += __expf(-d);                             // v_exp_f32
    }

    out[(size_t)b * OUT_W + D_SZ + k] = acc + bias[k];
}

// ---------------------------------------------------------------------------
extern "C" void kernel_launch(void* const* d_in, const int* in_sizes, int n_in,
                              void* d_out, int out_size, void* d_ws, size_t ws_size,
                              hipStream_t stream) {
    const float* x     = (const float*)d_in[0];   // [512, 2048]
    const float* theta = (const float*)d_in[1];   // [2048, 128, 5]
    const float* lws   = (const float*)d_in[2];   // [128, 5]
    const float* bias  = (const float*)d_in[3];   // [128]
    float* out = (float*)d_out;                   // [512, 2176]

    float* scale = (float*)d_ws;                  // 640 floats (padded to 1024)
    float* actv  = scale + 1024;                  // 512*640 floats (~1.31 MB)

    scale_kernel<<<(N_SZ + 255) / 256, 256, 0, stream>>>(theta, lws, scale);

    dim3 ggrid(N_SZ / 64, B_SZ / 16);             // (10, 32), 128 thr/block
    gemm_wmma_kernel<<<ggrid, 128, 0, stream>>>(x, theta, scale, actv);

    copy_x_kernel<<<(B_SZ * (D_SZ / 4) + 255) / 256, 256, 0, stream>>>(x, out);

    pairwise_kernel<<<K_SZ, 512, 0, stream>>>(actv, bias, out);
}